// StructAttentionLayer_3410204033350
// MI455X (gfx1250) — compile-verified
//
#include <hip/hip_runtime.h>
#include <math.h>

// GAT-style attention layer for MI455X (gfx1250, wave32).
// Bandwidth-bound (800 MB streamed once, ~1.2 GFLOP): single-pass design,
// rows staged in LDS, non-temporal B128 loads (stream > L2, zero reuse),
// attention-weighted aggregation done on the WMMA pipe (f32 16x16x4).

typedef float v2f __attribute__((ext_vector_type(2)));
typedef float v4f __attribute__((ext_vector_type(4)));
typedef float v8f __attribute__((ext_vector_type(8)));

#define ALPHA_LRELU 0.2f
#define NEG_INF_E  -1e30f
#define L_DIM 32
#define F_DIM 128
#define ROWS 3          // rows per workgroup: 3 * 16KB tile = 48KB LDS
#define THREADS 256     // 8 wave32s

__global__ __launch_bounds__(THREADS)
void gat_attention_kernel(const float* __restrict__ in,   // (M, 4096)
                          const float* __restrict__ a,    // (4224,)
                          float* __restrict__ out,        // (M, 128)
                          int M) {
  __shared__ float tile[ROWS * L_DIM * F_DIM];  // 48 KB row data
  __shared__ float d1[ROWS][L_DIM];             // rh . a1_l
  __shared__ float d2[ROWS][L_DIM];             // rh . a2
  __shared__ float sm[ROWS][L_DIM];             // sum_f rh (mask)
  __shared__ float att[ROWS][L_DIM];            // softmax weights
  __shared__ float hbuf[ROWS][F_DIM];           // aggregated features

  const int tid = threadIdx.x;
  const int m0  = (int)blockIdx.x * ROWS;

  // ---- Phase 1: stage ROWS rows into LDS with coalesced non-temporal B128 loads
  {
    const v4f* in4  = (const v4f*)in;
    v4f*       lds4 = (v4f*)tile;
    const long base4 = (long)m0 * (L_DIM * F_DIM / 4);
    const long lim4  = (long)M  * (L_DIM * F_DIM / 4);
#pragma unroll
    for (int i = 0; i < (ROWS * L_DIM * F_DIM / 4) / THREADS; ++i) {
      long g = base4 + i * THREADS + tid;
      v4f v = {0.f, 0.f, 0.f, 0.f};
      if (g < lim4) v = __builtin_nontemporal_load(in4 + g);   // TH=NT stream
      lds4[i * THREADS + tid] = v;
    }
  }
  __syncthreads();

  // ---- Phase 2: per (row, l) slice: dot(rh, a1_l), dot(rh, a2), sum(rh)
  // 8 lanes per (row,l) pair; 96 pairs in 3 sweeps of 32.
  {
    const int sub = tid & 7;
    const float* a2p = a + L_DIM * F_DIM;       // a2 = a[4096 + f]
#pragma unroll
    for (int s = 0; s < (ROWS * L_DIM * 8) / THREADS; ++s) {
      int p = s * (THREADS / 8) + (tid >> 3);   // 0..95
      int r = p >> 5;
      int l = p & (L_DIM - 1);
      const float* row = &tile[(r * L_DIM + l) * F_DIM];
      const float* a1p = a + l * F_DIM;         // a1 slice for this l
      float acc1 = 0.f, acc2 = 0.f, accs = 0.f;
#pragma unroll
      for (int j = 0; j < F_DIM / 8; ++j) {
        int f = sub + 8 * j;
        float v = row[f];
        acc1 = fmaf(v, a1p[f], acc1);
        acc2 = fmaf(v, a2p[f], acc2);
        accs += v;
      }
      // reduce across the 8 sub-lanes (contiguous within a wave32)
      acc1 += __shfl_xor(acc1, 1, 8); acc2 += __shfl_xor(acc2, 1, 8); accs += __shfl_xor(accs, 1, 8);
      acc1 += __shfl_xor(acc1, 2, 8); acc2 += __shfl_xor(acc2, 2, 8); accs += __shfl_xor(accs, 2, 8);
      acc1 += __shfl_xor(acc1, 4, 8); acc2 += __shfl_xor(acc2, 4, 8); accs += __shfl_xor(accs, 4, 8);
      if (sub == 0) {
        d1[r][l] = acc1;
        d2[r][l] = acc2;
        sm[r][l] = accs;
      }
    }
  }
  __syncthreads();

  // ---- Phase 3: masked leaky-relu + softmax over L=32 (3 threads, trivial work)
  if (tid < ROWS) {
    const int r = tid;
    float s_self = 0.f;
#pragma unroll
    for (int l = 0; l < L_DIM; ++l) s_self += d1[r][l];
    float e[L_DIM];
    float mx = -3.0e38f;
#pragma unroll
    for (int l = 0; l < L_DIM; ++l) {
      float z  = s_self + d2[r][l];
      float lr = (z >= 0.f) ? z : ALPHA_LRELU * z;
      float ev = (sm[r][l] != 0.f) ? lr : NEG_INF_E;
      e[l] = ev;
      mx = fmaxf(mx, ev);
    }
    float ssum = 0.f;
#pragma unroll
    for (int l = 0; l < L_DIM; ++l) {
      float pr = __expf(e[l] - mx);
      e[l] = pr;
      ssum += pr;
    }
    float inv = 1.f / ssum;
#pragma unroll
    for (int l = 0; l < L_DIM; ++l) att[r][l] = e[l] * inv;
  }
  __syncthreads();

  // ---- Phase 4: h[f] = sum_l att[l] * rh[l][f] on the matrix pipe.
  // Per row: D(16x16) += A(16x4) x B(4x16), A = RH^T f-tile (M dim = 16 f's),
  // att broadcast into all 16 B columns. Each wave owns one 16-wide f-tile.
  // ISA f32 A layout: lanes 0-15 hold K={0,1} in VGPR{0,1}; lanes 16-31 K={2,3}.
  {
    const int wave  = tid >> 5;
    const int lane  = tid & 31;
    const int fbase = wave * 16;            // 8 waves x 16 = 128 features
    const int half  = lane >> 4;            // 0 -> K pair {0,1}, 1 -> {2,3}
    const int fm    = fbase + (lane & 15);
    for (int r = 0; r < ROWS; ++r) {
      v8f c = {};
#pragma unroll
      for (int k0 = 0; k0 < L_DIM; k0 += 4) {
        int k = k0 + half * 2;
        v2f A, B;
        A.x = tile[(r * L_DIM + k)     * F_DIM + fm];
        A.y = tile[(r * L_DIM + k + 1) * F_DIM + fm];
        B.x = att[r][k];
        B.y = att[r][k + 1];
        c = __builtin_amdgcn_wmma_f32_16x16x4_f32(
                /*neg_a=*/false, A, /*neg_b=*/false, B,
                /*c_mod=*/(short)0, c, /*reuse_a=*/false, /*reuse_b=*/false);
      }
      // D layout: VGPR v, lanes 0-15 -> M=v, lanes 16-31 -> M=v+8; cols identical.
      if ((lane & 15) == 0) {
        int mb = fbase + (half ? 8 : 0);
#pragma unroll
        for (int v = 0; v < 8; ++v) hbuf[r][mb + v] = c[v];
      }
    }
  }
  __syncthreads();

  // ---- Phase 5: ELU-ish activation + coalesced store
  for (int idx = tid; idx < ROWS * F_DIM; idx += THREADS) {
    int r = idx >> 7;
    int f = idx & (F_DIM - 1);
    long m = (long)m0 + r;
    if (m < M) {
      float h = hbuf[r][f];
      out[m * F_DIM + f] = (h > 0.f) ? h : expm1f(h);
    }
  }
}

extern "C" void kernel_launch(void* const* d_in, const int* in_sizes, int n_in,
                              void* d_out, int out_size, void* d_ws, size_t ws_size,
                              hipStream_t stream) {
  const float* in = (const float*)d_in[0];   // (M, L*F) f32
  const float* a  = (const float*)d_in[1];   // (L*F + F,) f32
  float* out = (float*)d_out;                // (M, F) f32
  int M = in_sizes[0] / (L_DIM * F_DIM);
  int nwg = (M + ROWS - 1) / ROWS;
  gat_attention_kernel<<<dim3(nwg), dim3(THREADS), 0, stream>>>(in, a, out, M);
}